// ObjectBranch_31026843746481
// MI455X (gfx1250) — compile-verified
//
#include <hip/hip_runtime.h>
#include <hip/hip_bf16.h>
#include <climits>

#define GHT 2048
#define GWD 2048
#define KSEG 64
#define HID 128
#define NCOL 10
#define HEADS 4
#define HDIM 32
#define LN_EPS 1e-5f

#define AS1 __attribute__((address_space(1)))
#define AS3 __attribute__((address_space(3)))

typedef __attribute__((ext_vector_type(2))) float v2f;
typedef __attribute__((ext_vector_type(8))) float v8f;
typedef __attribute__((ext_vector_type(4))) int v4i;

#if __has_builtin(__builtin_amdgcn_wmma_f32_16x16x4_f32)
__device__ __forceinline__ v8f wmma_f32x4(v2f a, v2f b, v8f c) {
  // V_WMMA_F32_16X16X4_F32: D = A(16x4) * B(4x16) + C(16x16), all fp32
  return __builtin_amdgcn_wmma_f32_16x16x4_f32(false, a, false, b, (short)0, c,
                                               false, false);
}
#else
// Compile-only fallback (NOT layout-equivalent); histogram would show wmma==0.
__device__ __forceinline__ v8f wmma_f32x4(v2f a, v2f b, v8f c) {
  c[0] += a[0] * b[0] + a[1] * b[1];
  return c;
}
#endif

__device__ __forceinline__ void wait_async_all() {
#if __has_builtin(__builtin_amdgcn_s_wait_asynccnt)
  __builtin_amdgcn_s_wait_asynccnt(0);
#else
  asm volatile("s_wait_asynccnt 0" ::: "memory");
#endif
}

// Stage a 128x128 fp32 weight matrix (64 KB) global -> LDS.
// Preferred path: GLOBAL_LOAD_ASYNC_TO_LDS_B128 (ASYNCcnt-tracked DMA-style
// copy, no VGPR round trip); fallback: vectorized synchronous copy.
__device__ __forceinline__ void stage_weights(const float* __restrict__ Wg,
                                              float* __restrict__ wlds,
                                              int tid) {
#if __has_builtin(__builtin_amdgcn_global_load_async_to_lds_b128)
  #pragma unroll
  for (int j = 0; j < 16; ++j) {
    const int idx4 = (j * 256 + tid) * 4;  // 16-byte granules
    __builtin_amdgcn_global_load_async_to_lds_b128(
        (v4i*)(Wg + idx4), (AS3 v4i*)(wlds + idx4), 0, 0);
  }
  wait_async_all();
#else
  for (int i = tid * 4; i < HID * HID; i += 256 * 4) {
    *(float4*)(wlds + i) = *(const float4*)(Wg + i);
  }
#endif
}

// ---------------------------------------------------------------------------
// Workspace layout (int32 units):
//   [0 .. 639]   hist[64][10]
//   [640 .. 703] ymin[64]   [704 .. 767] ymax[64]
//   [768 .. 831] xmin[64]   [832 .. 895] xmax[64]
// ---------------------------------------------------------------------------

__global__ __launch_bounds__(256) void init_ws_kernel(int* __restrict__ ws) {
  for (int i = threadIdx.x; i < 896; i += 256) {
    int v;
    if (i < 640)      v = 0;         // hist
    else if (i < 704) v = INT_MAX;   // ymin
    else if (i < 768) v = INT_MIN;   // ymax
    else if (i < 832) v = INT_MAX;   // xmin
    else              v = INT_MIN;   // xmax
    ws[i] = v;
  }
}

// Bandwidth-bound segmented reduction: 32 cells/thread via int4 (B128) loads.
// A thread's 32 cells never cross a row (2048 % 32 == 0): row is constant,
// columns increase monotonically, so min/max per cached label collapse to
// first/last non-background column. Color counts pack into a 6-bit-per-color
// u64 (max 32 per color < 63), avoiding dynamically indexed register arrays.
__global__ __launch_bounds__(256) void seg_reduce_kernel(
    const int* __restrict__ grid, const int* __restrict__ labels,
    int* __restrict__ ws) {
  __shared__ int sh_hist[KSEG * NCOL];
  __shared__ int sh_ymin[KSEG], sh_ymax[KSEG], sh_xmin[KSEG], sh_xmax[KSEG];
  const int tid = threadIdx.x;
  for (int i = tid; i < KSEG * NCOL; i += 256) sh_hist[i] = 0;
  if (tid < KSEG) {
    sh_ymin[tid] = INT_MAX; sh_ymax[tid] = INT_MIN;
    sh_xmin[tid] = INT_MAX; sh_xmax[tid] = INT_MIN;
  }
  __syncthreads();

  const long long gtid = (long long)blockIdx.x * 256 + tid;
  const long long base = gtid * 32;
  if (base < (long long)GHT * GWD) {
    const int r  = (int)(base >> 11);   // base / 2048 (row, constant)
    const int c0 = (int)(base & 2047);  // base % 2048

    int cur = -1, cmin = 0, cmax = 0;
    unsigned long long cnt = 0;
    auto flush = [&]() {
      #pragma unroll
      for (int c = 0; c < NCOL; ++c) {
        const int v = (int)((cnt >> (6 * c)) & 63ull);
        if (v) atomicAdd(&sh_hist[cur * NCOL + c], v);
      }
      atomicMin(&sh_ymin[cur], r);
      atomicMax(&sh_ymax[cur], r);
      atomicMin(&sh_xmin[cur], cmin);
      atomicMax(&sh_xmax[cur], cmax);
    };

    for (int ch = 0; ch < 4; ++ch) {
      const long long off = base + ch * 8;
      const int4 l0 = *(const int4*)(labels + off);
      const int4 l1 = *(const int4*)(labels + off + 4);
      const int4 g0 = *(const int4*)(grid + off);
      const int4 g1 = *(const int4*)(grid + off + 4);
      const int labs[8] = {l0.x, l0.y, l0.z, l0.w, l1.x, l1.y, l1.z, l1.w};
      const int cols[8] = {g0.x, g0.y, g0.z, g0.w, g1.x, g1.y, g1.z, g1.w};
      #pragma unroll
      for (int j = 0; j < 8; ++j) {
        const int lab = labs[j];
        if (lab < 0 || lab >= KSEG) continue;  // background label == K
        const int cc = c0 + ch * 8 + j;
        if (lab != cur) {
          if (cur >= 0) flush();
          cur = lab; cnt = 0; cmin = cmax = cc;
        } else {
          cmax = cc;
        }
        cnt += 1ull << (cols[j] * 6);
      }
    }
    if (cur >= 0) flush();
  }
  __syncthreads();

  for (int i = tid; i < KSEG * NCOL; i += 256) {
    const int v = sh_hist[i];
    if (v) atomicAdd(&ws[i], v);
  }
  if (tid < KSEG && sh_ymin[tid] != INT_MAX) {
    atomicMin(&ws[640 + tid], sh_ymin[tid]);
    atomicMax(&ws[704 + tid], sh_ymax[tid]);
    atomicMin(&ws[768 + tid], sh_xmin[tid]);
    atomicMax(&ws[832 + tid], sh_xmax[tid]);
  }
}

// C[64][128] = A[64][128] @ W^T + bias   (C[i][j] = sum_k A[i][k] * W[j*128+k])
// W is async-staged into LDS first, then 4x8 tiles of 16x16 with 32 K-steps of
// V_WMMA_F32_16X16X4_F32, 4 tiles per wave (8 waves).
// ISA layouts: A lane m=lane%16, vgpr v holds K = 4s + 2*(lane/16) + v;
// B symmetric with n=lane%16; D element (v,lane) -> row v + 8*(lane/16).
__device__ __forceinline__ void wmma_mm_64x128x128(
    const float* __restrict__ Ash, const float* __restrict__ Wg,
    float* __restrict__ wlds, const float* __restrict__ bias,
    float* __restrict__ Csh, int tid) {
  __syncthreads();                 // prior readers of wlds are done
  stage_weights(Wg, wlds, tid);    // issues async copies + waits own ASYNCcnt
  __syncthreads();                 // all waves' staged data visible

  const int lane = tid & 31, wave = tid >> 5;
  const int hf = lane >> 4, l16 = lane & 15;
  for (int t = wave; t < 32; t += 8) {
    const int mt = t >> 3, nt = t & 7;
    const float* arow = Ash + (mt * 16 + l16) * HID;
    const float* wrow = wlds + (nt * 16 + l16) * HID;
    v8f acc = {};
    #pragma unroll 8
    for (int s = 0; s < 32; ++s) {
      const int kb = 4 * s + 2 * hf;
      const v2f a = *(const v2f*)(arow + kb);
      const v2f b = *(const v2f*)(wrow + kb);
      acc = wmma_f32x4(a, b, acc);
    }
    #pragma unroll
    for (int v = 0; v < 8; ++v) {
      const int row = mt * 16 + v + 8 * hf;
      const int col = nt * 16 + l16;
      Csh[row * HID + col] = acc[v] + bias[col];
    }
  }
}

__global__ __launch_bounds__(256) void final_kernel(
    const int* __restrict__ ws, const float* __restrict__ emb,
    const float* __restrict__ geo_w1, const float* __restrict__ geo_b1,
    const float* __restrict__ geo_w2, const float* __restrict__ geo_b2,
    const float* __restrict__ query, const float* __restrict__ wqkv,
    const float* __restrict__ bqkv, const float* __restrict__ wo,
    const float* __restrict__ bo, const float* __restrict__ ln_w,
    const float* __restrict__ ln_b, float* __restrict__ out) {
  __shared__ __align__(16) float feats[KSEG * HID];  // 32 KB
  __shared__ __align__(16) float buf[KSEG * HID];    // 32 KB (hidden/hist/kp/vp)
  __shared__ __align__(16) float wlds[HID * HID];    // 64 KB weight stage
  __shared__ __align__(16) float geo_in[KSEG * 4];
  __shared__ float countsf[KSEG];
  __shared__ float qv[HID], qp[HID], sc[HEADS * KSEG], ocat[HID], xbuf[HID];

  const int tid = threadIdx.x;
  const int lane = tid & 31, wave = tid >> 5;
  const int hf = lane >> 4, l16 = lane & 15;

  // --- geometry features + counts from workspace ---
  if (tid < KSEG) {
    int cnt = 0;
    for (int c = 0; c < NCOL; ++c) cnt += ws[tid * NCOL + c];
    const int ymin = ws[640 + tid], ymax = ws[704 + tid];
    const int xmin = ws[768 + tid], xmax = ws[832 + tid];
    float h = 0.f, w = 0.f, a = 0.f;
    if (cnt > 0) {
      h = (float)(ymax - ymin + 1);
      w = (float)(xmax - xmin + 1);
      a = (float)cnt * (1.0f / ((float)GHT * (float)GWD));
    }
    geo_in[tid * 4 + 0] = h;
    geo_in[tid * 4 + 1] = w;
    geo_in[tid * 4 + 2] = a;
    geo_in[tid * 4 + 3] = 0.f;
    countsf[tid] = cnt > 0 ? (float)cnt : 1.f;
  }
  if (tid < HID) qv[tid] = query[tid];
  __syncthreads();

  // --- hidden = relu(geo_in @ geo_w1^T + geo_b1); K=3 padded to 4 (1 step) ---
  for (int t = wave; t < 32; t += 8) {
    const int mt = t >> 3, nt = t & 7;
    const int m = mt * 16 + l16, n = nt * 16 + l16;
    const int kb = 2 * hf;
    const v2f a = *(const v2f*)&geo_in[m * 4 + kb];
    v2f b;  // clamped-index loads * mask: no divergent branches
    b.x = geo_w1[n * 3 + (kb < 2 ? kb : 2)] * (kb < 3 ? 1.f : 0.f);
    b.y = geo_w1[n * 3 + (kb + 1 < 2 ? kb + 1 : 2)] * (kb + 1 < 3 ? 1.f : 0.f);
    v8f acc = {};
    acc = wmma_f32x4(a, b, acc);
    #pragma unroll
    for (int v = 0; v < 8; ++v) {
      const int row = mt * 16 + v + 8 * hf, col = nt * 16 + l16;
      buf[row * HID + col] = fmaxf(acc[v] + geo_b1[col], 0.f);
    }
  }
  __syncthreads();

  // --- feats <- geo = hidden @ geo_w2^T + geo_b2 ---
  wmma_mm_64x128x128(buf, geo_w2, wlds, geo_b2, feats, tid);
  __syncthreads();

  // --- stage hist as fp32 [64][12] (K=10 padded to 12) into buf ---
  if (tid < KSEG) {
    for (int c = 0; c < 12; ++c)
      buf[tid * 12 + c] = (c < NCOL) ? (float)ws[tid * NCOL + c] : 0.f;
  }
  __syncthreads();

  // --- feats = 0.5 * (geo + (hist @ emb) / counts); 3 K-steps ---
  for (int t = wave; t < 32; t += 8) {
    const int mt = t >> 3, nt = t & 7;
    const int m = mt * 16 + l16, n = nt * 16 + l16;
    v8f acc = {};
    #pragma unroll
    for (int s = 0; s < 3; ++s) {
      const int kb = 4 * s + 2 * hf;
      const v2f a = *(const v2f*)&buf[m * 12 + kb];
      v2f b;
      b.x = emb[(kb < NCOL ? kb : NCOL - 1) * HID + n] * (kb < NCOL ? 1.f : 0.f);
      b.y = emb[(kb + 1 < NCOL ? kb + 1 : NCOL - 1) * HID + n] *
            (kb + 1 < NCOL ? 1.f : 0.f);
      acc = wmma_f32x4(a, b, acc);
    }
    #pragma unroll
    for (int v = 0; v < 8; ++v) {
      const int row = mt * 16 + v + 8 * hf, col = nt * 16 + l16;
      feats[row * HID + col] =
          0.5f * (feats[row * HID + col] + acc[v] / countsf[row]);
    }
  }
  __syncthreads();

  // --- 2 layers of single-query multihead cross-attention + layernorm ---
  const float scale = 0.17677669529663687f;  // 1/sqrt(32)
  for (int l = 0; l < 2; ++l) {
    const float* Wl = wqkv + l * 3 * HID * HID;
    const float* bl = bqkv + l * 3 * HID;
    // qp = q @ Wq^T + bq
    if (tid < HID) {
      const float* wr = Wl + tid * HID;
      float s = bl[tid];
      for (int k = 0; k < HID; ++k) s += qv[k] * wr[k];
      qp[tid] = s;
    }
    __syncthreads();
    // kp -> buf
    wmma_mm_64x128x128(feats, Wl + HID * HID, wlds, bl + HID, buf, tid);
    __syncthreads();
    // scores[h][k] = scale * <qp_h, kp_k_h>
    {
      const int h = tid >> 6, k = tid & 63;
      const float* kr = buf + k * HID + h * HDIM;
      const float* qr = qp + h * HDIM;
      float s = 0.f;
      for (int d = 0; d < HDIM; ++d) s += qr[d] * kr[d];
      sc[h * KSEG + k] = s * scale;
    }
    __syncthreads();
    // softmax per head (in place)
    if (tid < HEADS) {
      float mx = -1e30f;
      for (int k = 0; k < KSEG; ++k) mx = fmaxf(mx, sc[tid * KSEG + k]);
      float ssum = 0.f;
      for (int k = 0; k < KSEG; ++k) {
        const float e = __expf(sc[tid * KSEG + k] - mx);
        sc[tid * KSEG + k] = e;
        ssum += e;
      }
      const float inv = 1.f / ssum;
      for (int k = 0; k < KSEG; ++k) sc[tid * KSEG + k] *= inv;
    }
    __syncthreads();
    // vp -> buf (kp dead after scores)
    wmma_mm_64x128x128(feats, Wl + 2 * HID * HID, wlds, bl + 2 * HID, buf, tid);
    __syncthreads();
    // ocat[h*32+d] = sum_k attn[h][k] * vp[k][h*32+d]
    if (tid < HID) {
      const int h = tid >> 5;
      float s = 0.f;
      for (int k = 0; k < KSEG; ++k) s += sc[h * KSEG + k] * buf[k * HID + tid];
      ocat[tid] = s;
    }
    __syncthreads();
    // o = ocat @ wo^T + bo; residual
    if (tid < HID) {
      const float* wr = wo + l * HID * HID + tid * HID;
      float s = bo[l * HID + tid];
      for (int k = 0; k < HID; ++k) s += ocat[k] * wr[k];
      xbuf[tid] = qv[tid] + s;
    }
    __syncthreads();
    // layernorm
    if (tid < HID) {
      float mu = 0.f;
      for (int k = 0; k < HID; ++k) mu += xbuf[k];
      mu *= (1.f / HID);
      float var = 0.f;
      for (int k = 0; k < HID; ++k) {
        const float d = xbuf[k] - mu;
        var += d * d;
      }
      var *= (1.f / HID);
      qv[tid] = (xbuf[tid] - mu) * rsqrtf(var + LN_EPS) * ln_w[l * HID + tid] +
                ln_b[l * HID + tid];
    }
    __syncthreads();
  }

  if (tid < HID) out[tid] = qv[tid];
}

extern "C" void kernel_launch(void* const* d_in, const int* in_sizes, int n_in,
                              void* d_out, int out_size, void* d_ws,
                              size_t ws_size, hipStream_t stream) {
  const int* grid    = (const int*)d_in[0];
  const int* labels  = (const int*)d_in[1];
  const float* emb    = (const float*)d_in[2];
  const float* geo_w1 = (const float*)d_in[3];
  const float* geo_b1 = (const float*)d_in[4];
  const float* geo_w2 = (const float*)d_in[5];
  const float* geo_b2 = (const float*)d_in[6];
  const float* query  = (const float*)d_in[7];
  const float* wqkv   = (const float*)d_in[8];
  const float* bqkv   = (const float*)d_in[9];
  const float* wo     = (const float*)d_in[10];
  const float* bo     = (const float*)d_in[11];
  const float* ln_w   = (const float*)d_in[12];
  const float* ln_b   = (const float*)d_in[13];
  int* ws = (int*)d_ws;
  float* out = (float*)d_out;

  init_ws_kernel<<<1, 256, 0, stream>>>(ws);
  // 512 blocks * 256 threads * 32 cells/thread == 2048*2048 cells exactly
  seg_reduce_kernel<<<512, 256, 0, stream>>>(grid, labels, ws);
  final_kernel<<<1, 256, 0, stream>>>(ws, emb, geo_w1, geo_b1, geo_w2, geo_b2,
                                      query, wqkv, bqkv, wo, bo, ln_w, ln_b,
                                      out);
}